// YIQGNGCLoss_4337916969512
// MI455X (gfx1250) — compile-verified
//
#include <hip/hip_runtime.h>

// Fused 5x5 local-correlation (YIQGNGC channels==1 path) for MI455X gfx1250.
// One kernel, LDS-resident intermediates, async-to-LDS tile loads (ASYNCcnt),
// vertical 5-tap box reduction on the matrix pipe via V_WMMA_F32_16X16X4_F32.

#define OT    32            // output tile (32x32)
#define INR   40            // input tile rows/cols = OT + 8
#define XS    44            // LDS stride of input tiles (16B aligned rows)
#define HS    37            // stride of horizontal-sum arrays (36 cols used)
#define DS_   37            // stride of dx/dy arrays (36x36 used)
#define P2S   33            // stride of phase-2 horizontal-sum arrays (32 cols)
#define NPIX  (1016 * 1016)

typedef float v2f __attribute__((ext_vector_type(2)));
typedef float v8f __attribute__((ext_vector_type(8)));

// Vertical 5-tap box-sum of a 16-col strip as D = S(16x20) * H(20x16), using
// five chained V_WMMA_F32_16X16X4_F32 (K=20 in 5 chunks of 4).
// A layout (32-bit 16x4): lanes 0-15 hold K=0(v0),1(v1); lanes 16-31 K=2,3.
// B layout (32-bit 4x16): lanes 0-15 hold K=0(v0),1(v1) at N=lane;
//                         lanes 16-31 hold K=2,3 at N=lane-16.
// C/D layout: vgpr v holds M=v (lanes 0-15) / M=v+8 (lanes 16-31), N=lane%16.
__device__ inline v8f wmma_vbox5(const float* __restrict__ h, int i0b, int j0b,
                                 int lane) {
  const int m    = lane & 15;   // A row / also B col index for this lane
  const int half = lane >> 4;
  v8f acc = {0.f, 0.f, 0.f, 0.f, 0.f, 0.f, 0.f, 0.f};
#pragma unroll
  for (int q = 0; q < 5; ++q) {
    const int k0 = 4 * q + 2 * half;  // K index held in vgpr0 of this lane
    const int k1 = k0 + 1;            // K index held in vgpr1
    v2f a, b;
    a.x = (k0 >= m && k0 <= m + 4) ? 1.0f : 0.0f;  // banded selection S[m][k]
    a.y = (k1 >= m && k1 <= m + 4) ? 1.0f : 0.0f;
    b.x = h[(i0b + k0) * P2S + j0b + m];           // H[k][n], n == lane&15
    b.y = h[(i0b + k1) * P2S + j0b + m];
    acc = __builtin_amdgcn_wmma_f32_16x16x4_f32(false, a, false, b, (short)0,
                                                acc, false, false);
  }
  return acc;
}

__global__ __launch_bounds__(128) void gngc_kernel(
    const float* __restrict__ x, const float* __restrict__ y,
    float* __restrict__ out_corr, float* __restrict__ out_vv,
    float* __restrict__ out_c) {
  __shared__ __align__(16) float xt[INR * XS];
  __shared__ __align__(16) float yt[INR * XS];
  __shared__ float hxs[INR * HS];
  __shared__ float hys[INR * HS];
  __shared__ float dxs[36 * DS_];
  __shared__ float dys[36 * DS_];
  __shared__ float hc[36 * P2S];
  __shared__ float hvx[36 * P2S];
  __shared__ float hvy[36 * P2S];

  const int tid = threadIdx.x;
  const int img = blockIdx.z;
  const int gx0 = blockIdx.y * OT;  // tile origin row (input == output coords)
  const int gy0 = blockIdx.x * OT;  // tile origin col

  const float* xbase = x + (size_t)img * 1024 * 1024;
  const float* ybase = y + (size_t)img * 1024 * 1024;

  // ---- Phase 0: async tile loads (global -> LDS), b128 per lane ----------
  // 40 rows x 10 float4 per row per plane. OOB rows/cols clamped; clamped
  // cells only feed outputs >= 1016 which are never stored.
  for (int l = tid; l < INR * 10; l += 128) {
    const int rr = l / 10;
    const int kk = l - rr * 10;
    int r = gx0 + rr;      if (r > 1023) r = 1023;
    int c = gy0 + kk * 4;  if (c > 1020) c = 1020;
    const unsigned voff = (unsigned)((r << 10) + c) * 4u;
    const unsigned lx = (unsigned)(unsigned long long)&xt[rr * XS + kk * 4];
    const unsigned ly = (unsigned)(unsigned long long)&yt[rr * XS + kk * 4];
    asm volatile("global_load_async_to_lds_b128 %0, %1, %2"
                 :: "v"(lx), "v"(voff), "s"(xbase) : "memory");
    asm volatile("global_load_async_to_lds_b128 %0, %1, %2"
                 :: "v"(ly), "v"(voff), "s"(ybase) : "memory");
  }
  asm volatile("s_wait_asynccnt 0x0" ::: "memory");
  __syncthreads();

  // ---- Phase 1a: horizontal 5-sums of x,y (40 rows x 36 cols) ------------
  for (int l = tid; l < INR * 36; l += 128) {
    const int r = l / 36;
    const int b = l - r * 36;
    const float* px = &xt[r * XS + b];
    const float* py = &yt[r * XS + b];
    hxs[r * HS + b] = px[0] + px[1] + px[2] + px[3] + px[4];
    hys[r * HS + b] = py[0] + py[1] + py[2] + py[3] + py[4];
  }
  __syncthreads();

  // ---- Phase 1b: dx = center - boxmean (36 x 36) --------------------------
  for (int l = tid; l < 36 * 36; l += 128) {
    const int a = l / 36;
    const int b = l - a * 36;
    const float bx = hxs[a * HS + b] + hxs[(a + 1) * HS + b] +
                     hxs[(a + 2) * HS + b] + hxs[(a + 3) * HS + b] +
                     hxs[(a + 4) * HS + b];
    const float by = hys[a * HS + b] + hys[(a + 1) * HS + b] +
                     hys[(a + 2) * HS + b] + hys[(a + 3) * HS + b] +
                     hys[(a + 4) * HS + b];
    dxs[a * DS_ + b] = xt[(a + 2) * XS + b + 2] - bx * 0.04f;
    dys[a * DS_ + b] = yt[(a + 2) * XS + b + 2] - by * 0.04f;
  }
  __syncthreads();

  // ---- Phase 2a: horizontal 5-sums of dx*dy, dx^2, dy^2 (36 x 32) --------
  for (int l = tid; l < 36 * 32; l += 128) {
    const int a = l >> 5;
    const int j = l & 31;
    const float* pdx = &dxs[a * DS_ + j];
    const float* pdy = &dys[a * DS_ + j];
    float sc = 0.f, sxx = 0.f, syy = 0.f;
#pragma unroll
    for (int k = 0; k < 5; ++k) {
      const float ddx = pdx[k], ddy = pdy[k];
      sc  += ddx * ddy;
      sxx += ddx * ddx;
      syy += ddy * ddy;
    }
    hc[a * P2S + j]  = sc;
    hvx[a * P2S + j] = sxx;
    hvy[a * P2S + j] = syy;
  }
  __syncthreads();

  // ---- Phase 2b: vertical 5-sums on the matrix pipe + final math ---------
  // Each of the 4 waves owns one 16x16 output block; uniform control flow,
  // EXEC all ones at the WMMA sites.
  const int wave = tid >> 5;
  const int lane = tid & 31;
  const int i0b = (wave >> 1) << 4;
  const int j0b = (wave & 1) << 4;

  const v8f rc = wmma_vbox5(hc, i0b, j0b, lane);
  const v8f rx = wmma_vbox5(hvx, i0b, j0b, lane);
  const v8f ry = wmma_vbox5(hvy, i0b, j0b, lane);

  const int gj = gy0 + j0b + (lane & 15);
  const size_t pofs = (size_t)img * NPIX;
  if (gj < 1016) {
#pragma unroll
    for (int v = 0; v < 8; ++v) {
      const int gi = gx0 + i0b + v + 8 * (lane >> 4);
      if (gi < 1016) {
        const float c  = rc[v] * 0.04f;
        const float vx = rx[v] * 0.04f;
        const float vy = ry[v] * 0.04f;
        const float vv = sqrtf(vx * vy);
        const bool  low = vv < 1e-6f;
        const float cm  = low ? 0.0f : c;
        const float vvm = low ? 1e-6f : vv;
        const float corr = fminf(fmaxf(cm / vvm, 0.0f), 1.0f);
        const size_t o = pofs + (size_t)gi * 1016 + gj;
        out_corr[o] = corr;
        out_vv[o]   = vvm;
        out_c[o]    = cm;
      }
    }
  }
}

extern "C" void kernel_launch(void* const* d_in, const int* in_sizes, int n_in,
                              void* d_out, int out_size, void* d_ws,
                              size_t ws_size, hipStream_t stream) {
  const float* x = (const float*)d_in[0];
  const float* y = (const float*)d_in[1];
  // d_in[2] (mask) is unused by the reference's channels==1 path.
  float* out  = (float*)d_out;
  float* corr = out;                           // [8,1016,1016]
  float* vvp  = out + (size_t)8 * NPIX;        // [8,1016,1016]
  float* cp   = out + (size_t)16 * NPIX;       // [8,1016,1016]
  dim3 grid(32, 32, 8);  // (col tiles, row tiles, images)
  dim3 block(128);
  hipLaunchKernelGGL(gngc_kernel, grid, block, 0, stream, x, y, corr, vvp, cp);
}